// GraphEncoder_89043261981416
// MI455X (gfx1250) — compile-verified
//
#include <hip/hip_runtime.h>

// ---------------------------------------------------------------------------
// GCN 2-layer + mean-pool for MI455X (gfx1250, wave32, WMMA).
//  N=100000 nodes, E=1600000 edges, 64 graphs, dims 128 -> 256 -> 256.
//  GEMMs use V_WMMA_F32_16X16X4_F32 (exact fp32); graph scatter uses
//  L2-resident float atomics (feature table = 102 MB < 192 MB L2).
// ---------------------------------------------------------------------------

typedef __attribute__((ext_vector_type(2))) float v2f;
typedef __attribute__((ext_vector_type(8))) float v8f;

#define N_NODES   100000
#define N_EDGES   1600000
#define N_GRAPHS  64
#define IN_DIM    128
#define HID_DIM   256
#define OUT_DIM   256

// ---------------- utility -------------------------------------------------

__global__ void k_fill(float* __restrict__ p, float v, long n) {
  long i = (long)blockIdx.x * blockDim.x + threadIdx.x;
  long stride = (long)gridDim.x * blockDim.x;
  for (; i < n; i += stride) p[i] = v;
}

__global__ void k_count_deg(const int* __restrict__ cols, float* __restrict__ deg, int E) {
  int e = blockIdx.x * blockDim.x + threadIdx.x;
  if (e < E) atomicAdd(&deg[cols[e]], 1.0f);
}

__global__ void k_rsqrt_inplace(float* __restrict__ d, int n) {
  int i = blockIdx.x * blockDim.x + threadIdx.x;
  if (i < n) d[i] = rsqrtf(fmaxf(d[i], 1.0f));
}

// ---------------- fp32 WMMA GEMM ------------------------------------------
// T[m, n] = (sum_k X[m,k] * W[k,n]) * dinv[m]
// One wave computes one 16x16 tile via V_WMMA_F32_16X16X4_F32 (K-step 4).
// Block = 128 threads = 4 waves -> 16 rows x 64 cols per block.
// grid.x = M/16 (M = 100000, exact), grid.y = D/64.  K, D compile-time so
// every access is an immediate offset from one base address (24-bit IOFFSET).
//
// A operand (16x4 f32, 2 VGPRs): lane L: m = L%16, element v holds K = 2*(L/16)+v
// B operand (4x16 f32, 2 VGPRs): lane L: n = L%16, element v holds K = 2*(L/16)+v
// C/D (16x16 f32, 8 VGPRs):      lane L: n = L%16, VGPR v -> m = v + 8*(L/16)
template <int K, int D>
__global__ void k_gemm_wmma(const float* __restrict__ X,
                            const float* __restrict__ W,
                            const float* __restrict__ dinv,
                            float* __restrict__ T) {
  const int lane  = threadIdx.x & 31;
  const int wave  = threadIdx.x >> 5;
  const int lo    = lane & 15;
  const int hi    = lane >> 4;
  const int mbase = blockIdx.x * 16;
  const int nbase = (blockIdx.y * 4 + wave) * 16;

  // lane-private base pointers; all loop accesses become immediate offsets
  const float* __restrict__ xp = X + (long)(mbase + lo) * K + 2 * hi;
  const float* __restrict__ wp = W + (long)(2 * hi) * D + nbase + lo;

  v8f acc = {};
#pragma unroll
  for (int k0 = 0; k0 < K; k0 += 4) {
    v2f a = *(const v2f*)(xp + k0);      // A[m][k0+2hi], A[m][k0+2hi+1]
    v2f b;
    b.x = wp[(long)k0 * D];              // B[k0+2hi][n]
    b.y = wp[(long)(k0 + 1) * D];        // B[k0+2hi+1][n]
    acc = __builtin_amdgcn_wmma_f32_16x16x4_f32(
        /*neg_a=*/false, a, /*neg_b=*/false, b,
        /*c_mod=*/(short)0, acc, /*reuse_a=*/false, /*reuse_b=*/false);
  }

  const int mrow = mbase + 8 * hi;
  float* __restrict__ tp = T + (long)mrow * D + nbase + lo;
  const float* __restrict__ dv = dinv + mrow;
#pragma unroll
  for (int v = 0; v < 8; ++v) {
    tp[(long)v * D] = acc[v] * dv[v];    // row m = mrow + v
  }
}

// ---------------- edge scatter (message passing) --------------------------
// AGG[c, :] += T[r, :] * dinv[c]   (T already pre-scaled by dinv[r])
// One wave per edge; lanes cover 256 dims with float4 gathers.
__global__ void k_scatter(const float* __restrict__ T,
                          const int* __restrict__ rows,
                          const int* __restrict__ cols,
                          const float* __restrict__ dinv,
                          float* __restrict__ AGG, int E) {
  const int e = blockIdx.x * 8 + (threadIdx.x >> 5);
  if (e >= E) return;
  const int lane = threadIdx.x & 31;
  const int r = rows[e];
  const int c = cols[e];
  const float w = dinv[c];
  const float4* __restrict__ src = (const float4*)(T + (long)r * 256);
  float* __restrict__ dst = AGG + (long)c * 256;
#pragma unroll
  for (int it = 0; it < 2; ++it) {
    const int d4 = lane + it * 32;   // float4 index 0..63
    const float4 v = src[d4];
    const int d = d4 * 4;
    atomicAdd(dst + d + 0, v.x * w);
    atomicAdd(dst + d + 1, v.y * w);
    atomicAdd(dst + d + 2, v.z * w);
    atomicAdd(dst + d + 3, v.w * w);
  }
}

// ---------------- self-loop + bias + ReLU ---------------------------------
// AGG[i,d] = relu(AGG[i,d] + T[i,d]*dinv[i] + b[d]);  one block per node.
__global__ void k_finish(float* __restrict__ AGG,
                         const float* __restrict__ T,
                         const float* __restrict__ dinv,
                         const float* __restrict__ bias) {
  const int node = blockIdx.x;
  const int d = threadIdx.x;                 // blockDim = 256
  const long idx = (long)node * 256 + d;
  const float v = AGG[idx] + T[idx] * dinv[node] + bias[d];
  AGG[idx] = fmaxf(v, 0.0f);
}

// ---------------- global mean pool ----------------------------------------
__global__ void k_pool(const float* __restrict__ H,
                       const int* __restrict__ batch,
                       float* __restrict__ pool,
                       float* __restrict__ cnt, int N) {
  const int node = blockIdx.x * 8 + (threadIdx.x >> 5);
  if (node >= N) return;
  const int lane = threadIdx.x & 31;
  const int g = batch[node];
  if (lane == 0) atomicAdd(&cnt[g], 1.0f);
  const float4* __restrict__ src = (const float4*)(H + (long)node * 256);
  float* __restrict__ dst = pool + (long)g * 256;
#pragma unroll
  for (int it = 0; it < 2; ++it) {
    const int d4 = lane + it * 32;
    const float4 v = src[d4];
    const int d = d4 * 4;
    atomicAdd(dst + d + 0, v.x);
    atomicAdd(dst + d + 1, v.y);
    atomicAdd(dst + d + 2, v.z);
    atomicAdd(dst + d + 3, v.w);
  }
}

__global__ void k_divide(const float* __restrict__ pool,
                         const float* __restrict__ cnt,
                         float* __restrict__ out) {
  const int g = blockIdx.x;
  const int d = threadIdx.x;
  out[g * 256 + d] = pool[g * 256 + d] / fmaxf(cnt[g], 1.0f);
}

// ---------------- host launcher -------------------------------------------
extern "C" void kernel_launch(void* const* d_in, const int* in_sizes, int n_in,
                              void* d_out, int out_size, void* d_ws, size_t ws_size,
                              hipStream_t stream) {
  const float* x     = (const float*)d_in[0];
  const int*   ei    = (const int*)d_in[1];     // (2, E): rows then cols
  const int*   batch = (const int*)d_in[2];
  const float* W1    = (const float*)d_in[3];
  const float* b1    = (const float*)d_in[4];
  const float* W2    = (const float*)d_in[5];
  const float* b2    = (const float*)d_in[6];
  float* out = (float*)d_out;

  const int N = N_NODES;
  const int E = N_EDGES;
  const int* rows = ei;
  const int* cols = ei + E;

  // workspace layout (floats)
  float* ws = (float*)d_ws;
  const long off_dinv = 0;
  const long off_A    = 100096;                       // padded past N, 16B aligned
  const long off_B    = off_A + (long)N * 256;
  const long off_pool = off_B + (long)N * 256;
  const long off_cnt  = off_pool + (long)N_GRAPHS * 256;
  const long total    = off_cnt + N_GRAPHS;
  if (ws_size < (size_t)total * sizeof(float)) return;  // not enough scratch

  float* dinv = ws + off_dinv;
  float* bufA = ws + off_A;     // post-GEMM, pre-scaled features t
  float* bufB = ws + off_B;     // scatter accumulator / activations h
  float* pool = ws + off_pool;
  float* cnt  = ws + off_cnt;

  const long feat = (long)N * 256;

  // ---- degree & normalization (deg includes self-loop) ----
  k_fill<<<512, 256, 0, stream>>>(dinv, 1.0f, N);
  k_count_deg<<<(E + 255) / 256, 256, 0, stream>>>(cols, dinv, E);
  k_rsqrt_inplace<<<(N + 255) / 256, 256, 0, stream>>>(dinv, N);

  // ---- layer 1: t = (x @ W1) * dinv[row] ; agg = scatter ; relu(+self,+b) ----
  k_fill<<<2048, 256, 0, stream>>>(bufB, 0.0f, feat);
  k_gemm_wmma<IN_DIM, HID_DIM><<<dim3(N / 16, HID_DIM / 64), 128, 0, stream>>>(
      x, W1, dinv, bufA);
  k_scatter<<<(E + 7) / 8, 256, 0, stream>>>(bufA, rows, cols, dinv, bufB, E);
  k_finish<<<N, 256, 0, stream>>>(bufB, bufA, dinv, b1);   // bufB = h1

  // ---- layer 2 ----
  k_gemm_wmma<HID_DIM, OUT_DIM><<<dim3(N / 16, OUT_DIM / 64), 128, 0, stream>>>(
      bufB, W2, dinv, bufA);
  k_fill<<<2048, 256, 0, stream>>>(bufB, 0.0f, feat);      // after GEMM2 consumed h1
  k_scatter<<<(E + 7) / 8, 256, 0, stream>>>(bufA, rows, cols, dinv, bufB, E);
  k_finish<<<N, 256, 0, stream>>>(bufB, bufA, dinv, b2);   // bufB = h2

  // ---- mean pool over graphs ----
  k_fill<<<64, 256, 0, stream>>>(pool, 0.0f, (long)N_GRAPHS * 256 + N_GRAPHS);
  k_pool<<<(N + 7) / 8, 256, 0, stream>>>(bufB, batch, pool, cnt, N);
  k_divide<<<N_GRAPHS, 256, 0, stream>>>(pool, cnt, out);
}